// SelfAttention_4922032521376
// MI455X (gfx1250) — compile-verified
//
#include <hip/hip_runtime.h>
#include <hip/hip_bf16.h>
#include <math.h>

#define DDIM   2048
#define BATCH  32
#define KSPLIT 8
#define KCHUNK (DDIM / KSPLIT)   // 256
#define NTILES (DDIM / 16)       // 128
#define JBLK   256
#define LOG2E  1.4426950408889634f

typedef __attribute__((ext_vector_type(2))) float v2f;
typedef __attribute__((ext_vector_type(8))) float v8f;

// ---------------------------------------------------------------------------
// Kernel 1: QKV projection partial GEMM with fp32 WMMA (16x16x4).
// out = x @ W^T  as A[32,2048] x B[2048,2048], where B[k][n] = W[n][k]
// (so B columns are W rows -> contiguous along k, streamed over the kk loop).
// One wave (32 threads) per (matrix, 16-wide N tile, 256-wide K chunk).
// Both 16-row M tiles (rows 0-15 and 16-31) share one B fragment per step.
// ---------------------------------------------------------------------------
__global__ __launch_bounds__(32) void qkv_gemm_partial(
    const float* __restrict__ x,
    const float* __restrict__ Wq,
    const float* __restrict__ Wk,
    const float* __restrict__ Wv,
    float* __restrict__ partial)   // [3][KSPLIT][32][DDIM]
{
    int bid    = blockIdx.x;               // 3 * NTILES * KSPLIT blocks
    int kchunk = bid % KSPLIT;
    int tmp    = bid / KSPLIT;
    int ntile  = tmp % NTILES;
    int mat    = tmp / NTILES;

    const float* W = (mat == 0) ? Wq : ((mat == 1) ? Wk : Wv);

    int lane   = threadIdx.x;              // 0..31 (wave32)
    int half   = lane >> 4;                // 0: lanes 0-15, 1: lanes 16-31
    int lidx   = lane & 15;
    int lane_k = half * 2;                 // A/B VGPR0 holds K=0 (lo) / K=2 (hi)
    int n0     = ntile * 16;
    int k0base = kchunk * KCHUNK;

    // A fragment rows (x is [32, DDIM] row-major)
    const float* xa0 = x + (size_t)lidx * DDIM;          // M tile 0: rows 0-15
    const float* xa1 = x + (size_t)(lidx + 16) * DDIM;   // M tile 1: rows 16-31
    // B fragment column n = W row n (contiguous in k)
    const float* wb  = W + (size_t)(n0 + lidx) * DDIM;

    v8f c0 = {};
    v8f c1 = {};
    for (int kk = 0; kk < KCHUNK; kk += 4) {
        int k0 = k0base + kk + lane_k;
        v2f a0 = { xa0[k0], xa0[k0 + 1] };   // A[16x4]: {K, K+1} per lane half
        v2f a1 = { xa1[k0], xa1[k0 + 1] };
        v2f b  = { wb[k0],  wb[k0 + 1]  };   // B[4x16] mirrored layout
        c0 = __builtin_amdgcn_wmma_f32_16x16x4_f32(
                 false, a0, false, b, (short)0, c0, false, false);
        c1 = __builtin_amdgcn_wmma_f32_16x16x4_f32(
                 false, a1, false, b, (short)0, c1, false, false);
    }

    // C/D layout: VGPR r -> row (r + half*8), col = n0 + lidx
    float* p    = partial + (((size_t)mat * KSPLIT + kchunk) * BATCH) * DDIM;
    int    row0 = half * 8;
    int    col  = n0 + lidx;
#pragma unroll
    for (int r = 0; r < 8; ++r) {
        p[(size_t)(row0 + r)      * DDIM + col] = c0[r];
        p[(size_t)(row0 + r + 16) * DDIM + col] = c1[r];
    }
}

// ---------------------------------------------------------------------------
// Kernel 2: reduce K-split partials + bias -> q, k, v  [32, DDIM] each
// ---------------------------------------------------------------------------
__global__ __launch_bounds__(256) void qkv_reduce(
    const float* __restrict__ partial,
    const float* __restrict__ bq,
    const float* __restrict__ bk,
    const float* __restrict__ bv,
    float* __restrict__ q,
    float* __restrict__ k,
    float* __restrict__ v)
{
    int idx = blockIdx.x * blockDim.x + threadIdx.x;   // 3*BATCH*DDIM threads
    int mat = idx / (BATCH * DDIM);
    int rem = idx % (BATCH * DDIM);
    int row = rem / DDIM;
    int col = rem % DDIM;

    float s = 0.0f;
#pragma unroll
    for (int kc = 0; kc < KSPLIT; ++kc)
        s += partial[(((size_t)mat * KSPLIT + kc) * BATCH + row) * DDIM + col];

    const float* bias = (mat == 0) ? bq : ((mat == 1) ? bk : bv);
    float*       dst  = (mat == 0) ? q  : ((mat == 1) ? k  : v);
    dst[(size_t)row * DDIM + col] = s + bias[col];
}

// ---------------------------------------------------------------------------
// Kernel 3: per (b, j): m_j = (k_j>=0 ? qmax : qmin)*k_j,
//                      Z_j = sum_i exp(q_i*k_j - m_j),  c_j = v_j / Z_j
// One block per (b, 256-wide j chunk); q row staged in LDS; block max/min.
// Inner loop in exp2 domain (log2e folded into k_j/m_j once per thread),
// 4 independent accumulators to keep v_exp_f32 (TRANS) issue-limited.
// ---------------------------------------------------------------------------
__global__ __launch_bounds__(256) void softmax_denom(
    const float* __restrict__ q,
    const float* __restrict__ k,
    const float* __restrict__ v,
    float* __restrict__ mbuf,
    float* __restrict__ cbuf)
{
    __shared__ float qs[DDIM];
    __shared__ float smax[256];
    __shared__ float smin[256];

    int b   = blockIdx.x / (DDIM / JBLK);
    int jc  = blockIdx.x % (DDIM / JBLK);
    int tid = threadIdx.x;

    float lmax = -INFINITY, lmin = INFINITY;
    for (int i = tid; i < DDIM; i += 256) {
        float qv = q[(size_t)b * DDIM + i];
        qs[i] = qv;
        lmax = fmaxf(lmax, qv);
        lmin = fminf(lmin, qv);
    }
    smax[tid] = lmax;
    smin[tid] = lmin;
    __syncthreads();
    for (int s = 128; s > 0; s >>= 1) {
        if (tid < s) {
            smax[tid] = fmaxf(smax[tid], smax[tid + s]);
            smin[tid] = fminf(smin[tid], smin[tid + s]);
        }
        __syncthreads();
    }
    float qmax = smax[0];
    float qmin = smin[0];

    int   j   = jc * JBLK + tid;
    float kj  = k[(size_t)b * DDIM + j];
    float m   = (kj >= 0.0f ? qmax : qmin) * kj;   // natural-domain stabilizer
    float kj2 = kj * LOG2E;                        // exp2-domain coefficients
    float m2  = m * LOG2E;

    float z0 = 0.0f, z1 = 0.0f, z2 = 0.0f, z3 = 0.0f;
    for (int i = 0; i < DDIM; i += 4) {
        z0 += __builtin_amdgcn_exp2f(qs[i + 0] * kj2 - m2);
        z1 += __builtin_amdgcn_exp2f(qs[i + 1] * kj2 - m2);
        z2 += __builtin_amdgcn_exp2f(qs[i + 2] * kj2 - m2);
        z3 += __builtin_amdgcn_exp2f(qs[i + 3] * kj2 - m2);
    }
    float Z = (z0 + z1) + (z2 + z3);

    mbuf[(size_t)b * DDIM + j] = m;
    cbuf[(size_t)b * DDIM + j] = v[(size_t)b * DDIM + j] / Z;
}

// ---------------------------------------------------------------------------
// Kernel 4: out[b,i] = sum_j exp(q_i*k_j - m_j) * c_j
// One block per (b, 256-wide i chunk); k/m/c tiled through LDS with log2e
// pre-folded at staging. 4 accumulators; inner body = v_fma + v_exp + v_fmac.
// ---------------------------------------------------------------------------
__global__ __launch_bounds__(256) void attn_out(
    const float* __restrict__ q,
    const float* __restrict__ k,
    const float* __restrict__ mbuf,
    const float* __restrict__ cbuf,
    float* __restrict__ out)
{
    __shared__ float ks[JBLK];
    __shared__ float ms[JBLK];
    __shared__ float cs[JBLK];

    int b   = blockIdx.x / (DDIM / JBLK);
    int ic  = blockIdx.x % (DDIM / JBLK);
    int tid = threadIdx.x;
    int i   = ic * JBLK + tid;

    float qi = q[(size_t)b * DDIM + i];
    float a0 = 0.0f, a1 = 0.0f, a2 = 0.0f, a3 = 0.0f;

    for (int j0 = 0; j0 < DDIM; j0 += JBLK) {
        __syncthreads();
        ks[tid] = k[(size_t)b * DDIM + j0 + tid] * LOG2E;     // exp2 domain
        ms[tid] = mbuf[(size_t)b * DDIM + j0 + tid] * LOG2E;
        cs[tid] = cbuf[(size_t)b * DDIM + j0 + tid];
        __syncthreads();
#pragma unroll 4
        for (int jj = 0; jj < JBLK; jj += 4) {
            a0 += __builtin_amdgcn_exp2f(qi * ks[jj + 0] - ms[jj + 0]) * cs[jj + 0];
            a1 += __builtin_amdgcn_exp2f(qi * ks[jj + 1] - ms[jj + 1]) * cs[jj + 1];
            a2 += __builtin_amdgcn_exp2f(qi * ks[jj + 2] - ms[jj + 2]) * cs[jj + 2];
            a3 += __builtin_amdgcn_exp2f(qi * ks[jj + 3] - ms[jj + 3]) * cs[jj + 3];
        }
    }
    out[(size_t)b * DDIM + i] = (a0 + a1) + (a2 + a3);
}

// ---------------------------------------------------------------------------
// Launch
// ---------------------------------------------------------------------------
extern "C" void kernel_launch(void* const* d_in, const int* in_sizes, int n_in,
                              void* d_out, int out_size, void* d_ws, size_t ws_size,
                              hipStream_t stream) {
    const float* x  = (const float*)d_in[0];
    const float* Wq = (const float*)d_in[1];
    const float* bq = (const float*)d_in[2];
    const float* Wk = (const float*)d_in[3];
    const float* bk = (const float*)d_in[4];
    const float* Wv = (const float*)d_in[5];
    const float* bv = (const float*)d_in[6];
    float* out = (float*)d_out;

    char* ws = (char*)d_ws;
    // workspace layout (bytes)
    const size_t PARTIAL_BYTES = (size_t)3 * KSPLIT * BATCH * DDIM * sizeof(float); // 6 MB
    const size_t ROW_BYTES     = (size_t)BATCH * DDIM * sizeof(float);              // 256 KB
    float* partial = (float*)(ws);
    float* q       = (float*)(ws + PARTIAL_BYTES);
    float* k       = (float*)(ws + PARTIAL_BYTES + 1 * ROW_BYTES);
    float* v       = (float*)(ws + PARTIAL_BYTES + 2 * ROW_BYTES);
    float* mbuf    = (float*)(ws + PARTIAL_BYTES + 3 * ROW_BYTES);
    float* cbuf    = (float*)(ws + PARTIAL_BYTES + 4 * ROW_BYTES);

    // 1) WMMA partial GEMMs: one wave per (matrix, ntile, kchunk)
    qkv_gemm_partial<<<3 * NTILES * KSPLIT, 32, 0, stream>>>(x, Wq, Wk, Wv, partial);

    // 2) reduce partials + bias
    qkv_reduce<<<(3 * BATCH * DDIM) / 256, 256, 0, stream>>>(partial, bq, bk, bv, q, k, v);

    // 3) per-(b,j) stabilizer + denominator
    softmax_denom<<<BATCH * (DDIM / JBLK), 256, 0, stream>>>(q, k, v, mbuf, cbuf);

    // 4) weighted sum over j
    attn_out<<<BATCH * (DDIM / JBLK), 256, 0, stream>>>(q, k, mbuf, cbuf, out);
}